// ChamferDist_76252849373403
// MI455X (gfx1250) — compile-verified
//
#include <hip/hip_runtime.h>

typedef float v2f __attribute__((ext_vector_type(2)));
typedef float v8f __attribute__((ext_vector_type(8)));

#define CB 8
#define CN 4096
#define CM 4096
#define INF_BITS 0x7F800000

static __device__ __forceinline__ int imin2(int a, int b) { return a < b ? a : b; }

// Pack src into (x, y, z, ||s||^2) float4s, init per-source mins and per-batch sums.
__global__ __launch_bounds__(256) void chamfer_prep(const float* __restrict__ src,
                                                    float4* __restrict__ srcPad,
                                                    int* __restrict__ srcMin,
                                                    float* __restrict__ compSum) {
  int i = blockIdx.x * blockDim.x + threadIdx.x;
  if (i < CB * CM) {
    float x = src[3 * i], y = src[3 * i + 1], z = src[3 * i + 2];
    srcPad[i] = make_float4(x, y, z, x * x + y * y + z * z);
    srcMin[i] = INF_BITS;
  }
  if (i < CB) compSum[i] = 0.0f;
}

__global__ __launch_bounds__(256) void chamfer_tiles(const float* __restrict__ tar,
                                                     const float4* __restrict__ srcPad,
                                                     int* __restrict__ srcMin,
                                                     float* __restrict__ compSum) {
  const int b    = blockIdx.y;
  const int wave = threadIdx.x >> 5;   // 8 wave32s per workgroup
  const int lane = threadIdx.x & 31;
  const int col  = lane & 15;
  const int hi   = lane >> 4;          // half-wave: K=0,1 vs K=2,3 operand slots
  const int tbase = blockIdx.x * 128 + wave * 16;

  __shared__ int lsMin[CM];            // 16 KB per-source running min (i32-ordered fp32)
  for (int i = threadIdx.x; i < CM; i += 256) lsMin[i] = INF_BITS;
  __syncthreads();

  const float*  tb = tar + (size_t)b * CN * 3;
  const float4* sb = srcPad + (size_t)b * CM;

  // A tile (16 targets x K=4): rows = targets, K slots = (-2x, -2y | -2z, 1).
  // With B columns = (x, y | z, ||s||^2), WMMA yields -2 t.s + ||s||^2.
  v2f amat;
  {
    const float* tp = tb + (tbase + col) * 3;
    float x = tp[0], y = tp[1], z = tp[2];
    amat.x = hi ? (-2.0f * z) : (-2.0f * x);
    amat.y = hi ? 1.0f        : (-2.0f * y);
  }

  // C accumulator: c[r] = ||t||^2 of C-matrix row r + 8*hi  -> WMMA output is
  // the full squared distance ||t||^2 + ||s||^2 - 2 t.s.
  v8f cinit;
#pragma unroll
  for (int r = 0; r < 8; ++r) {
    const float* q = tb + (tbase + r + 8 * hi) * 3;
    float qx = q[0], qy = q[1], qz = q[2];
    cinit[r] = qx * qx + qy * qy + qz * qz;
  }

  int minT[8];
#pragma unroll
  for (int r = 0; r < 8; ++r) minT[r] = INF_BITS;

  // Half-waves swap source tiles per commit so each ds_min instruction hits
  // 32 distinct LDS words (conflict-free across the 64 banks).
  const int c0off = hi * 16 + col;
  const int c1off = (hi ^ 1) * 16 + col;

  // 4 source tiles (64 sources) per iteration: 4 independent b128 loads in
  // flight, 4 WMMAs sharing the A operand, fixed costs amortized 4x.
  for (int mt = 0; mt < CM; mt += 64) {
    float4 p0 = sb[mt + col];
    float4 p1 = sb[mt + 16 + col];
    float4 p2 = sb[mt + 32 + col];
    float4 p3 = sb[mt + 48 + col];

    v2f b0, b1, b2, b3;
    b0.x = hi ? p0.z : p0.x;  b0.y = hi ? p0.w : p0.y;
    b1.x = hi ? p1.z : p1.x;  b1.y = hi ? p1.w : p1.y;
    b2.x = hi ? p2.z : p2.x;  b2.y = hi ? p2.w : p2.y;
    b3.x = hi ? p3.z : p3.x;  b3.y = hi ? p3.w : p3.y;

    v8f c0 = __builtin_amdgcn_wmma_f32_16x16x4_f32(false, amat, false, b0, (short)0, cinit, false, false);
    v8f c1 = __builtin_amdgcn_wmma_f32_16x16x4_f32(false, amat, false, b1, (short)0, cinit, false, false);
    v8f c2 = __builtin_amdgcn_wmma_f32_16x16x4_f32(false, amat, false, b2, (short)0, cinit, false, false);
    v8f c3 = __builtin_amdgcn_wmma_f32_16x16x4_f32(false, amat, false, b3, (short)0, cinit, false, false);

    // Signed-i32 bit compares: correct order for all finite d >= 0; tiny
    // negative cancellation artifacts (true d^2 ~ 0) sort below all positives
    // and are clamped to 0 once at the end -> no per-element clamp needed.
    int m0 = INF_BITS, m1 = INF_BITS, m2 = INF_BITS, m3 = INF_BITS;
#pragma unroll
    for (int r = 0; r < 8; ++r) {
      int d0 = __float_as_int(c0[r]);
      int d1 = __float_as_int(c1[r]);
      int d2 = __float_as_int(c2[r]);
      int d3 = __float_as_int(c3[r]);
      minT[r] = imin2(minT[r], imin2(imin2(d0, d1), imin2(d2, d3)));
      m0 = imin2(m0, d0);
      m1 = imin2(m1, d1);
      m2 = imin2(m2, d2);
      m3 = imin2(m3, d3);
    }
    // Commit each half-wave's partial (8-row) column min directly: ds_min_i32
    // without return is fire-and-forget -> no bpermute, no in-loop DS waits.
    // The partner half's 8 rows reach the same word via the swapped commit.
    atomicMin(&lsMin[mt + c0off],      hi ? m1 : m0);
    atomicMin(&lsMin[mt + c1off],      hi ? m0 : m1);
    atomicMin(&lsMin[mt + 32 + c0off], hi ? m3 : m2);
    atomicMin(&lsMin[mt + 32 + c1off], hi ? m2 : m3);
  }

  // per-target (row) min across the 16 columns within each half-wave
#pragma unroll
  for (int r = 0; r < 8; ++r) {
    int v = minT[r];
    v = imin2(v, __shfl_xor(v, 1, 32));
    v = imin2(v, __shfl_xor(v, 2, 32));
    v = imin2(v, __shfl_xor(v, 4, 32));
    v = imin2(v, __shfl_xor(v, 8, 32));
    minT[r] = v;
  }
  if (col == 0) {  // lane 0 (rows 0-7) and lane 16 (rows 8-15)
    float s = 0.0f;
#pragma unroll
    for (int r = 0; r < 8; ++r) {
      int v = minT[r] < 0 ? 0 : minT[r];   // deferred clamp of cancellation artifacts
      s += sqrtf(__int_as_float(v));
    }
    atomicAdd(&compSum[b], s);
  }

  __syncthreads();
  for (int i = threadIdx.x; i < CM; i += 256)
    atomicMin(&srcMin[(size_t)b * CM + i], lsMin[i]);
}

__global__ __launch_bounds__(256) void chamfer_finalize(const int* __restrict__ srcMin,
                                                        const float* __restrict__ compSum,
                                                        float* __restrict__ out) {
  __shared__ float red[256];
  float accMean = 0.0f;
  for (int b = 0; b < CB; ++b) {
    float s = 0.0f;
    for (int i = threadIdx.x; i < CM; i += 256) {
      int v = srcMin[b * CM + i];
      v = v < 0 ? 0 : v;                   // deferred clamp
      s += sqrtf(__int_as_float(v));
    }
    red[threadIdx.x] = s;
    __syncthreads();
    for (int off = 128; off > 0; off >>= 1) {
      if (threadIdx.x < off) red[threadIdx.x] += red[threadIdx.x + off];
      __syncthreads();
    }
    if (threadIdx.x == 0) accMean += red[0] / (float)CM;
    __syncthreads();
  }
  if (threadIdx.x == 0) {
    float compMean = 0.0f;
    for (int b = 0; b < CB; ++b) compMean += compSum[b] / (float)CN;
    float accuracy = accMean / (float)CB;
    float complete = compMean / (float)CB;
    out[0] = accuracy;                     // accuracy
    out[1] = complete;                     // complete
    out[2] = 0.5f * (accuracy + complete); // chamfer
  }
}

extern "C" void kernel_launch(void* const* d_in, const int* in_sizes, int n_in,
                              void* d_out, int out_size, void* d_ws, size_t ws_size,
                              hipStream_t stream) {
  const float* tar = (const float*)d_in[0];
  const float* src = (const float*)d_in[1];
  float* out = (float*)d_out;
  // ws layout: srcMin i32[B*M] (128 KB) | compSum f32[B] | srcPad float4[B*M] (512 KB)
  int* srcMin = (int*)d_ws;
  float* compSum = (float*)((char*)d_ws + (size_t)CB * CM * sizeof(int));
  float4* srcPad = (float4*)((char*)d_ws + (size_t)CB * CM * sizeof(int) + 32);

  chamfer_prep<<<(CB * CM + 255) / 256, 256, 0, stream>>>(src, srcPad, srcMin, compSum);
  chamfer_tiles<<<dim3(CN / 128, CB), 256, 0, stream>>>(tar, srcPad, srcMin, compSum);
  chamfer_finalize<<<1, 256, 0, stream>>>(srcMin, compSum, out);
}